// TopkSampler_51496657879110
// MI455X (gfx1250) — compile-verified
//
#include <hip/hip_runtime.h>

#define TPB   1024      // 32 waves (wave32) -> one full WGP
#define NCAND 2048      // top-k candidate capacity (K=900 plus ties margin)

typedef unsigned int       u32;
typedef unsigned long long u64;

// ---- gfx1250 async global->LDS copy (ASYNCcnt path, CDNA5 ISA) ----
__device__ __forceinline__ void async_load_b128_to_lds(const void* gptr, const void* ldsptr) {
    // Generic shared pointers carry the LDS byte offset in their low 32 bits
    // (aperture base lives entirely in the upper DWORD).
    u32 laddr = (u32)(size_t)ldsptr;
    u64 gaddr = (u64)(size_t)gptr;
    asm volatile("global_load_async_to_lds_b128 %0, %1, off"
                 :: "v"(laddr), "v"(gaddr) : "memory");
}
__device__ __forceinline__ void wait_async_le1() {
    asm volatile("s_wait_asynccnt 0x1" ::: "memory");
}
__device__ __forceinline__ void wait_async_le0() {
    asm volatile("s_wait_asynccnt 0x0" ::: "memory");
}

// LDS: union overlays the pass-1 staging buffers with the later-phase buffers.
union __align__(16) ShMem {
    float stage[2][TPB * 4];                      // 32 KB double-buffer (float4/thread)
    struct {
        u64 cand[NCAND];                          // 16 KB candidate (key<<32 | ~idx)
        u32 whist[16][256];                       // 16 KB per-wave-pair histograms
    } l;
};

__global__ __launch_bounds__(TPB) void topk_softmax_calib(
    const float* __restrict__ sal,   // (B, HW)
    const float* __restrict__ msk,   // (HW)
    float* __restrict__ out,         // [B*K scores | B*K indices | B*HW calib]
    int B, int HW, int K)
{
    const int b   = blockIdx.x;
    const int tid = threadIdx.x;
    const int wp  = tid >> 6;                 // wave-pair id (wave32: 2 waves share a hist)
    const int nIter = HW / (4 * TPB);         // 64 float4-iterations per row

    const float4* salRow = (const float4*)(sal + (size_t)b * HW);
    const float4* msk4   = (const float4*)msk;
    float4* calib4 = (float4*)(out + (size_t)2 * B * K + (size_t)b * HW);

    __shared__ ShMem sh;
    __shared__ float red[TPB];
    __shared__ u32   hfin[256];
    __shared__ float sM, sInvSum;
    __shared__ u32   sPrefix, sCntGt, sNC;

    // logits*log2e:  y = (sal+mask)*(1/T)*log2(e);  softmax == 2^(y-M)/sum 2^(y-M)
    const float C = 2.0f * 1.44269504088896340736f;

    // ---------- Pass 1: row max (async global->LDS double-buffered stream) ----------
    float m = -3.0e38f;
    async_load_b128_to_lds(salRow + tid, &sh.stage[0][tid * 4]);
    for (int it = 0; it < nIter; ++it) {
        if (it + 1 < nIter) {
            async_load_b128_to_lds(salRow + (it + 1) * TPB + tid,
                                   &sh.stage[(it + 1) & 1][tid * 4]);
            wait_async_le1();   // loads complete in order: buffer (it&1) is ready
        } else {
            wait_async_le0();
        }
        const float4 v  = *reinterpret_cast<const float4*>(&sh.stage[it & 1][tid * 4]);
        const float4 mv = msk4[it * TPB + tid];
        m = fmaxf(m, fmaxf(fmaxf((v.x + mv.x) * C, (v.y + mv.y) * C),
                           fmaxf((v.z + mv.z) * C, (v.w + mv.w) * C)));
    }
    red[tid] = m; __syncthreads();
    for (int off = TPB / 2; off > 0; off >>= 1) {
        if (tid < off) red[tid] = fmaxf(red[tid], red[tid + off]);
        __syncthreads();
    }
    if (tid == 0) sM = red[0];
    __syncthreads();
    const float M = sM;

    // ---------- Pass 2: sum of 2^(y - M)  (L2-resident re-read) ----------
    float s = 0.f;
    for (int it = 0; it < nIter; ++it) {
        const float4 v  = salRow[it * TPB + tid];
        const float4 mv = msk4[it * TPB + tid];
        s += __builtin_amdgcn_exp2f((v.x + mv.x) * C - M);
        s += __builtin_amdgcn_exp2f((v.y + mv.y) * C - M);
        s += __builtin_amdgcn_exp2f((v.z + mv.z) * C - M);
        s += __builtin_amdgcn_exp2f((v.w + mv.w) * C - M);
    }
    __syncthreads();
    red[tid] = s; __syncthreads();
    for (int off = TPB / 2; off > 0; off >>= 1) {
        if (tid < off) red[tid] += red[tid + off];
        __syncthreads();
    }
    if (tid == 0) sInvSum = 1.0f / red[0];
    __syncthreads();
    const float invS = sInvSum;

    // ---------- Pass 3: write calibrated map + MSB-byte histogram ----------
    for (int i = tid; i < 16 * 256; i += TPB) ((u32*)sh.l.whist)[i] = 0u;
    __syncthreads();
    for (int it = 0; it < nIter; ++it) {
        const float4 v  = salRow[it * TPB + tid];
        const float4 mv = msk4[it * TPB + tid];
        float4 o;
        o.x = __builtin_amdgcn_exp2f((v.x + mv.x) * C - M) * invS;
        o.y = __builtin_amdgcn_exp2f((v.y + mv.y) * C - M) * invS;
        o.z = __builtin_amdgcn_exp2f((v.z + mv.z) * C - M) * invS;
        o.w = __builtin_amdgcn_exp2f((v.w + mv.w) * C - M) * invS;
        calib4[it * TPB + tid] = o;
        atomicAdd(&sh.l.whist[wp][__float_as_uint(o.x) >> 24], 1u);
        atomicAdd(&sh.l.whist[wp][__float_as_uint(o.y) >> 24], 1u);
        atomicAdd(&sh.l.whist[wp][__float_as_uint(o.z) >> 24], 1u);
        atomicAdd(&sh.l.whist[wp][__float_as_uint(o.w) >> 24], 1u);
    }
    __syncthreads();
    if (tid < 256) {
        u32 t = 0;
        for (int w = 0; w < 16; ++w) t += sh.l.whist[w][tid];
        hfin[tid] = t;
    }
    __syncthreads();
    if (tid == 0) {
        u32 cg = 0; int bin = 255;
        for (; bin > 0; --bin) { u32 c = hfin[bin]; if (cg + c >= (u32)K) break; cg += c; }
        sPrefix = (u32)bin; sCntGt = cg;
    }
    __syncthreads();

    // ---------- Passes 4-6: refine radix digits 1..3 on the written soft values ----------
    for (int d = 1; d <= 3; ++d) {
        for (int i = tid; i < 16 * 256; i += TPB) ((u32*)sh.l.whist)[i] = 0u;
        __syncthreads();
        const u32 pfx = sPrefix;
        const u32 hiShift  = 32u - 8u * (u32)d;
        const u32 curShift = 24u - 8u * (u32)d;
        for (int it = 0; it < nIter; ++it) {
            const float4 o = calib4[it * TPB + tid];   // block-coherent after __syncthreads
            u32 k0 = __float_as_uint(o.x);
            u32 k1 = __float_as_uint(o.y);
            u32 k2 = __float_as_uint(o.z);
            u32 k3 = __float_as_uint(o.w);
            if ((k0 >> hiShift) == pfx) atomicAdd(&sh.l.whist[wp][(k0 >> curShift) & 255u], 1u);
            if ((k1 >> hiShift) == pfx) atomicAdd(&sh.l.whist[wp][(k1 >> curShift) & 255u], 1u);
            if ((k2 >> hiShift) == pfx) atomicAdd(&sh.l.whist[wp][(k2 >> curShift) & 255u], 1u);
            if ((k3 >> hiShift) == pfx) atomicAdd(&sh.l.whist[wp][(k3 >> curShift) & 255u], 1u);
        }
        __syncthreads();
        if (tid < 256) {
            u32 t = 0;
            for (int w = 0; w < 16; ++w) t += sh.l.whist[w][tid];
            hfin[tid] = t;
        }
        __syncthreads();
        if (tid == 0) {
            u32 cg = sCntGt; int bin = 255;
            for (; bin > 0; --bin) { u32 c = hfin[bin]; if (cg + c >= (u32)K) break; cg += c; }
            sPrefix = (pfx << 8) | (u32)bin; sCntGt = cg;
        }
        __syncthreads();
    }
    const u32 T = sPrefix;   // exact K-th largest softmax key

    // ---------- Pass 7: compact candidates (key >= T) ----------
    if (tid == 0) sNC = 0u;
    for (int i = tid; i < NCAND; i += TPB) sh.l.cand[i] = 0ull;
    __syncthreads();
    for (int it = 0; it < nIter; ++it) {
        const float4 o = calib4[it * TPB + tid];
        const u32 base = (u32)((it * TPB + tid) * 4);
        u32 kk[4] = { __float_as_uint(o.x), __float_as_uint(o.y),
                      __float_as_uint(o.z), __float_as_uint(o.w) };
        #pragma unroll
        for (int c = 0; c < 4; ++c) {
            if (kk[c] >= T) {
                u32 slot = atomicAdd(&sNC, 1u);
                if (slot < NCAND)
                    sh.l.cand[slot] = ((u64)kk[c] << 32) | (u32)(~(base + (u32)c));
            }
        }
    }
    __syncthreads();

    // ---------- Bitonic sort, descending by (key, -index) ----------
    for (u32 span = 2; span <= NCAND; span <<= 1) {
        for (u32 j = span >> 1; j > 0; j >>= 1) {
            for (u32 i = tid; i < NCAND; i += TPB) {
                const u32 l = i ^ j;
                if (l > i) {
                    const u64 a = sh.l.cand[i], c = sh.l.cand[l];
                    const bool up = ((i & span) == 0);
                    if (up ? (a < c) : (a > c)) { sh.l.cand[i] = c; sh.l.cand[l] = a; }
                }
            }
            __syncthreads();
        }
    }

    // ---------- Emit top-K scores and indices ----------
    float* scores = out + (size_t)b * K;
    float* idxs   = out + (size_t)B * K + (size_t)b * K;
    for (int j = tid; j < K; j += TPB) {
        const u64 c = sh.l.cand[j];
        scores[j] = __uint_as_float((u32)(c >> 32));
        idxs[j]   = (float)(~(u32)c);   // indices < 2^18: exact in fp32
    }
}

extern "C" void kernel_launch(void* const* d_in, const int* in_sizes, int n_in,
                              void* d_out, int out_size, void* d_ws, size_t ws_size,
                              hipStream_t stream) {
    (void)n_in; (void)d_ws; (void)ws_size;
    const float* sal = (const float*)d_in[0];
    const float* msk = (const float*)d_in[1];
    const int HW = in_sizes[1];                       // 262144
    const int B  = in_sizes[0] / HW;                  // 128
    const int K  = (int)(((long long)out_size - (long long)B * HW) / (2LL * B)); // 900
    topk_softmax_calib<<<dim3(B), dim3(TPB), 0, stream>>>(sal, msk, (float*)d_out, B, HW, K);
}